// FullPairwise_48241072668760
// MI455X (gfx1250) — compile-verified
//
#include <hip/hip_runtime.h>
#include <hip/hip_bf16.h>
#include <math.h>

typedef float v2f __attribute__((ext_vector_type(2)));
typedef float v8f __attribute__((ext_vector_type(8)));

#define NATOMS   2048
#define NMOL     4
#define CUT2     (5.2f * 5.2f)
#define TILES    (NATOMS / 16)                        // 128
#define TILE_PAIRS (TILES * (TILES + 1) / 2)          // 8256 (ti <= tj)
#define P_PAIRS  ((size_t)NATOMS * (NATOMS - 1) / 2)  // 2,096,128
#define MPTOT    ((size_t)NMOL * P_PAIRS)             // 8,384,512

// base offset in the triangular tile-pair list for tile-row a
__device__ __forceinline__ int tbase(int a) {
    return a * TILES - a * (a - 1) / 2;
}

// One wave32 per (molecule, 16x16 tile-pair). D = A*B via v_wmma_f32_16x16x4_f32
// with A row m = (-2x, -2y, -2z, |xi|^2), B col n = (x, y, z, 1):
//   D[m][n] = |xi|^2 - 2 xi.xj ;  d2 = D[m][n] + |xj|^2 (lane-resident add)
__global__ void __launch_bounds__(256) pairwise_tiles_kernel(
    const int*   __restrict__ species,
    const float* __restrict__ coords,
    float*       __restrict__ out)
{
    const int lane = threadIdx.x & 31;
    const int w    = blockIdx.x * 8 + (threadIdx.x >> 5);   // global wave id
    const int m    = w / TILE_PAIRS;                        // molecule
    const int t    = w % TILE_PAIRS;                        // tile-pair (uniform per wave)

    // Invert triangular tile-pair index: t = tbase(ti) + (tj - ti)
    float s  = sqrtf((float)(257 * 257 - 8 * t));
    int   ti = (int)((257.0f - s) * 0.5f);
    if (ti < 0) ti = 0;
    if (ti >= TILES) ti = TILES - 1;
    while (ti > 0 && tbase(ti) > t) --ti;
    while (ti + 1 < TILES && tbase(ti + 1) <= t) ++ti;
    const int tj = ti + (t - tbase(ti));

    const int    col   = lane & 15;
    const int    half  = lane >> 4;          // 0: K=0,1  1: K=2,3
    const size_t mbase = (size_t)m * NATOMS;

    // ---- B operand: j-atoms (columns), col = lane%16 ----
    const int    ja = tj * 16 + col;
    const float* cj = coords + (mbase + (size_t)ja) * 3;
    float bx = cj[0], by = cj[1], bz = cj[2];
    if (species[mbase + ja] == -1) { bx = by = bz = __builtin_nanf(""); }
    const float nj = bx * bx + by * by + bz * bz;

    // ---- A operand: i-atoms (rows), row = lane%16 ----
    const int    ia = ti * 16 + col;
    const float* ci = coords + (mbase + (size_t)ia) * 3;
    float ax = ci[0], ay = ci[1], az = ci[2];
    if (species[mbase + ia] == -1) { ax = ay = az = __builtin_nanf(""); }
    const float ni = ax * ax + ay * ay + az * az;

    v2f A, B;
    if (half == 0) { A[0] = -2.0f * ax; A[1] = -2.0f * ay; B[0] = bx; B[1] = by;   }
    else           { A[0] = -2.0f * az; A[1] = ni;         B[0] = bz; B[1] = 1.0f; }

    v8f c = {};
    // (neg_a, A, neg_b, B, c_mod, C, reuse_a, reuse_b)
    c = __builtin_amdgcn_wmma_f32_16x16x4_f32(false, A, false, B, (short)0, c,
                                              false, false);

    // Output regions (all written as float, concatenated in return order)
    float* idx0    = out;                 // atom_index12 row 0: [M*P]
    float* idx1    = out + MPTOT;         // atom_index12 row 1: [M*P]
    float* maskout = out + 5 * MPTOT;     // in-cutoff mask:     [M*P]

    const int j = tj * 16 + col;
    #pragma unroll
    for (int d = 0; d < 8; ++d) {
        const int i = ti * 16 + d + half * 8;   // C/D layout: VGPR d -> row d / d+8
        if (i < j) {
            const float d2 = c[d] + nj;
            const int   tri = i * (2 * NATOMS - 1 - i) / 2 + (j - i - 1);
            const size_t p  = (size_t)m * P_PAIRS + (size_t)tri;
            idx0[p]    = (float)(i + m * NATOMS);
            idx1[p]    = (float)(j + m * NATOMS);
            maskout[p] = (d2 <= CUT2) ? 1.0f : 0.0f;   // NaN -> false, as reference
        }
    }
}

// shift_values region is all zeros: [2*MPTOT, 5*MPTOT), 16B-aligned, multiple of 4
__global__ void zero_shifts_kernel(float* __restrict__ out)
{
    float4* p = (float4*)(out + 2 * MPTOT);
    const size_t n4 = (3 * MPTOT) / 4;
    for (size_t idx = (size_t)blockIdx.x * blockDim.x + threadIdx.x; idx < n4;
         idx += (size_t)gridDim.x * blockDim.x) {
        p[idx] = make_float4(0.0f, 0.0f, 0.0f, 0.0f);
    }
}

extern "C" void kernel_launch(void* const* d_in, const int* in_sizes, int n_in,
                              void* d_out, int out_size, void* d_ws, size_t ws_size,
                              hipStream_t stream) {
    const int*   species = (const int*)d_in[0];
    const float* coords  = (const float*)d_in[1];
    float*       out     = (float*)d_out;

    // 100 MB of zeros: grid-stride float4 stores
    zero_shifts_kernel<<<2048, 256, 0, stream>>>(out);

    // 4 molecules * 8256 tile-pairs = 33,024 waves; 8 waves/block -> 4128 blocks
    pairwise_tiles_kernel<<<(NMOL * TILE_PAIRS) / 8, 256, 0, stream>>>(
        species, coords, out);
}